// AttentionShare_52106543235148
// MI455X (gfx1250) — compile-verified
//
#include <hip/hip_runtime.h>
#include <math.h>
#include <stdint.h>

#define B_ 32
#define S_ 2048
#define D_ 1024
#define SCALE 0.03125f      // 1/sqrt(A=1024)
#define LN_EPS 1e-5f
#define NSB 16              // S-blocks for ctx partial accumulation
#define SB (S_ / NSB)       // 128 rows per block

typedef __attribute__((ext_vector_type(2))) float v2f;
typedef __attribute__((ext_vector_type(8))) float v8f;

// ---------------------------------------------------------------------------
// Y[M,N] = act( X[M,K] @ W[N,K]^T ), one wave per 16x16 tile, wmma f32 16x16x4
// A layout (ISA 7.12.2): lane l -> row m=l%16; VGPR0/1 hold K = 2*(l/16) + {0,1}
// B fed as columns: lane l -> col n=l%16; VGPR0/1 hold K = 2*(l/16) + {0,1}
// D layout: 8 VGPRs, lane l -> col n=l%16, VGPR v -> row m = v + 8*(l/16)
// ---------------------------------------------------------------------------
__global__ void gemm_xwt_wmma(const float* __restrict__ X,
                              const float* __restrict__ W,
                              float* __restrict__ Y,
                              int M, int N, int K, int do_tanh)
{
    int wave = (blockIdx.x * blockDim.x + threadIdx.x) >> 5;
    int lane = threadIdx.x & 31;
    int ntiles = N >> 4;
    int total  = (M >> 4) * ntiles;
    if (wave >= total) return;                 // wave-uniform: EXEC stays all-1s
    int m0 = (wave / ntiles) << 4;
    int n0 = (wave % ntiles) << 4;
    int r = lane & 15;
    int h = lane >> 4;

    const float* xa = X + (size_t)(m0 + r) * K + 2 * h;
    const float* wb = W + (size_t)(n0 + r) * K + 2 * h;

    v8f acc = {};
    for (int k0 = 0; k0 < K; k0 += 4) {
        v2f a = *(const v2f*)(xa + k0);
        v2f b = *(const v2f*)(wb + k0);
        acc = __builtin_amdgcn_wmma_f32_16x16x4_f32(
            /*neg_a=*/false, a, /*neg_b=*/false, b,
            /*c_mod=*/(short)0, acc, /*reuse_a=*/false, /*reuse_b=*/false);
    }
#pragma unroll
    for (int v = 0; v < 8; ++v) {
        int m = m0 + v + 8 * h;
        float val = acc[v];
        if (do_tanh) val = tanhf(val);
        Y[(size_t)m * N + n0 + r] = val;
    }
}

// Y[M,N] = X[M,K] @ W[K,N]   (W not transposed; column loads coalesce over lanes)
__global__ void gemm_xw_wmma(const float* __restrict__ X,
                             const float* __restrict__ W,
                             float* __restrict__ Y,
                             int M, int N, int K)
{
    int wave = (blockIdx.x * blockDim.x + threadIdx.x) >> 5;
    int lane = threadIdx.x & 31;
    int ntiles = N >> 4;
    int total  = (M >> 4) * ntiles;
    if (wave >= total) return;
    int m0 = (wave / ntiles) << 4;
    int n0 = (wave % ntiles) << 4;
    int r = lane & 15;
    int h = lane >> 4;

    const float* xa = X + (size_t)(m0 + r) * K + 2 * h;

    v8f acc = {};
    for (int k0 = 0; k0 < K; k0 += 4) {
        v2f a = *(const v2f*)(xa + k0);
        v2f b;
        b.x = W[(size_t)(k0 + 2 * h)     * N + n0 + r];
        b.y = W[(size_t)(k0 + 2 * h + 1) * N + n0 + r];
        acc = __builtin_amdgcn_wmma_f32_16x16x4_f32(
            false, a, false, b, (short)0, acc, false, false);
    }
#pragma unroll
    for (int v = 0; v < 8; ++v) {
        int m = m0 + v + 8 * h;
        Y[(size_t)m * N + n0 + r] = acc[v];
    }
}

// logits[b,s] = SCALE * dot(meta[b,s,:], qk[b,:]) — one wave per (b,s) row.
// All 8 waves of a block share the same batch b (256 blocks per batch exactly),
// so qk[b] (4 KB) is staged into LDS once per block with an async global->LDS
// copy (ASYNCcnt path), then consumed via DS reads.
__global__ void logits_kernel(const float* __restrict__ meta,
                              const float* __restrict__ qk,
                              float* __restrict__ logits)
{
    __shared__ float qlds[D_];
    int t    = threadIdx.x;
    int gt   = blockIdx.x * blockDim.x + t;
    int wave = gt >> 5;
    int lane = t & 31;
    int b = wave / S_;
    int s = wave % S_;

    // ---- async stage qk[b] into LDS: 256 lanes x 16B = 4 KB ----
    {
        int bblk = (blockIdx.x * 8) / S_;      // block-uniform batch index
        unsigned     ldsa = (unsigned)(uintptr_t)&qlds[4 * t];
        const float* gsrc = qk + (size_t)bblk * D_ + 4 * t;
        asm volatile("global_load_async_to_lds_b128 %0, %1, off"
                     :: "v"(ldsa), "v"(gsrc) : "memory");
        asm volatile("s_wait_asynccnt 0" ::: "memory");
    }
    __syncthreads();

    const float4* row = (const float4*)(meta + ((size_t)b * S_ + s) * D_);
    const float4* qv  = (const float4*)qlds;
    float acc = 0.0f;
#pragma unroll
    for (int j = 0; j < D_ / 128; ++j) {        // 8 iters: lane-coalesced float4
        float4 mv = row[lane + 32 * j];
        float4 qq = qv[lane + 32 * j];
        acc += mv.x * qq.x + mv.y * qq.y + mv.z * qq.z + mv.w * qq.w;
    }
    for (int off = 16; off > 0; off >>= 1)
        acc += __shfl_down(acc, off, 32);
    if (lane == 0) logits[(size_t)b * S_ + s] = acc * SCALE;
}

// softmax over S per batch; writes normalized weight straight into d_out region
__global__ void softmax_kernel(const float* __restrict__ logits,
                               float* __restrict__ weight)
{
    __shared__ float red[256];
    int b = blockIdx.x, t = threadIdx.x;
    const float* lr = logits + (size_t)b * S_;
    float v[8];
    float m = -3.4e38f;
#pragma unroll
    for (int j = 0; j < 8; ++j) { v[j] = lr[t + 256 * j]; m = fmaxf(m, v[j]); }
    red[t] = m; __syncthreads();
    for (int o = 128; o > 0; o >>= 1) {
        if (t < o) red[t] = fmaxf(red[t], red[t + o]);
        __syncthreads();
    }
    float rowmax = red[0]; __syncthreads();
    float s = 0.0f;
#pragma unroll
    for (int j = 0; j < 8; ++j) { v[j] = expf(v[j] - rowmax); s += v[j]; }
    red[t] = s; __syncthreads();
    for (int o = 128; o > 0; o >>= 1) {
        if (t < o) red[t] += red[t + o];
        __syncthreads();
    }
    float inv = 1.0f / red[0];
    float* wr = weight + (size_t)b * S_;
#pragma unroll
    for (int j = 0; j < 8; ++j) wr[t + 256 * j] = v[j] * inv;
}

// ctx partial: ctx_p[sblk,b,:] = sum_{s in block} weight[b,s] * meta[b,s,:]
__global__ void ctx_partial_kernel(const float* __restrict__ meta,
                                   const float* __restrict__ weight,
                                   float* __restrict__ ctx_p)
{
    int b = blockIdx.x, sblk = blockIdx.y, t = threadIdx.x;
    const float4* mb = (const float4*)(meta + ((size_t)b * S_ + (size_t)sblk * SB) * D_);
    const float*  w  = weight + (size_t)b * S_ + (size_t)sblk * SB;
    float4 acc = make_float4(0.f, 0.f, 0.f, 0.f);
    for (int s = 0; s < SB; ++s) {
        __builtin_prefetch(&mb[(size_t)(s + 4) * (D_ / 4) + t], 0, 0);
        float  ws = w[s];
        float4 mv = mb[(size_t)s * (D_ / 4) + t];
        acc.x += ws * mv.x; acc.y += ws * mv.y;
        acc.z += ws * mv.z; acc.w += ws * mv.w;
    }
    float4* cp = (float4*)(ctx_p + (size_t)(sblk * B_ + b) * D_);
    cp[t] = acc;
}

// deterministic ordered reduce of the NSB partials
__global__ void ctx_reduce_kernel(const float* __restrict__ ctx_p,
                                  float* __restrict__ ctx)
{
    int i = blockIdx.x * 256 + threadIdx.x;
    float a = 0.0f;
#pragma unroll
    for (int p = 0; p < NSB; ++p) a += ctx_p[(size_t)p * B_ * D_ + i];
    ctx[i] = a;
}

// LayerNorm over D per batch row
__global__ void ln_kernel(const float* __restrict__ H,
                          const float* __restrict__ lnw,
                          const float* __restrict__ lnb,
                          float* __restrict__ out)
{
    __shared__ float red[256];
    int b = blockIdx.x, t = threadIdx.x;
    const float* hr = H + (size_t)b * D_;
    float v[4];
    float s = 0.0f;
#pragma unroll
    for (int j = 0; j < 4; ++j) { v[j] = hr[t + 256 * j]; s += v[j]; }
    red[t] = s; __syncthreads();
    for (int o = 128; o > 0; o >>= 1) { if (t < o) red[t] += red[t + o]; __syncthreads(); }
    float mu = red[0] * (1.0f / D_); __syncthreads();
    float vs = 0.0f;
#pragma unroll
    for (int j = 0; j < 4; ++j) { float d = v[j] - mu; vs += d * d; }
    red[t] = vs; __syncthreads();
    for (int o = 128; o > 0; o >>= 1) { if (t < o) red[t] += red[t + o]; __syncthreads(); }
    float var = red[0] * (1.0f / D_);
    float inv = rsqrtf(var + LN_EPS);
#pragma unroll
    for (int j = 0; j < 4; ++j) {
        int d = t + 256 * j;
        out[(size_t)b * D_ + d] = (v[j] - mu) * inv * lnw[d] + lnb[d];
    }
}

extern "C" void kernel_launch(void* const* d_in, const int* in_sizes, int n_in,
                              void* d_out, int out_size, void* d_ws, size_t ws_size,
                              hipStream_t stream)
{
    const float* meta   = (const float*)d_in[0];   // [B,S,Dv]
    const float* hidden = (const float*)d_in[1];   // [B,Dk]
    const float* WK     = (const float*)d_in[2];   // [A,Dv]
    const float* WQ     = (const float*)d_in[3];   // [A,Dk]
    const float* WV     = (const float*)d_in[4];   // [Dv,Dv]
    const float* WO     = (const float*)d_in[5];   // [Dv,Dv]
    const float* ln_w   = (const float*)d_in[6];
    const float* ln_b   = (const float*)d_in[7];

    float* out    = (float*)d_out;                 // [B,Dv] first output
    float* weight = out + B_ * D_;                 // [B,S,1] second output

    float* ws     = (float*)d_ws;
    float* q      = ws;                            // 32768
    float* qk     = ws + 32768;                    // 32768
    float* logits = ws + 65536;                    // 65536
    float* ctx_p  = ws + 131072;                   // NSB*B*D = 524288
    float* ctx    = ws + 655360;                   // 32768
    float* att    = ws + 688128;                   // 32768
    float* hbuf   = ws + 720896;                   // 32768

    // q = hidden @ WQ^T                                 [32,1024]
    gemm_xwt_wmma<<<32, 128, 0, stream>>>(hidden, WQ, q, B_, D_, D_, 0);
    // qk = q @ WK   (folded K-projection)               [32,1024]
    gemm_xw_wmma<<<32, 128, 0, stream>>>(q, WK, qk, B_, D_, D_);
    // logits[b,s] = scale * meta[b,s,:].qk[b,:]   (pass 1 over meta, 256 MB)
    logits_kernel<<<(B_ * S_ * 32) / 256, 256, 0, stream>>>(meta, qk, logits);
    // softmax over S -> weight output
    softmax_kernel<<<B_, 256, 0, stream>>>(logits, weight);
    // ctx partials (pass 2 over meta, 256 MB) + deterministic reduce
    ctx_partial_kernel<<<dim3(B_, NSB), 256, 0, stream>>>(meta, weight, ctx_p);
    ctx_reduce_kernel<<<(B_ * D_) / 256, 256, 0, stream>>>(ctx_p, ctx);
    // att = ctx @ WV^T   (folded V-projection)
    gemm_xwt_wmma<<<32, 128, 0, stream>>>(ctx, WV, att, B_, D_, D_, 0);
    // h = tanh(att @ WO^T)
    gemm_xwt_wmma<<<32, 128, 0, stream>>>(att, WO, hbuf, B_, D_, D_, 1);
    // LayerNorm -> out
    ln_kernel<<<B_, 256, 0, stream>>>(hbuf, ln_w, ln_b, out);
}